// PYG_GIN_50697793962364
// MI455X (gfx1250) — compile-verified
//
#include <hip/hip_runtime.h>

typedef __attribute__((ext_vector_type(2))) float v2f;
typedef __attribute__((ext_vector_type(8))) float v8f;

constexpr int NN = 10000;   // nodes
constexpr int NE = 640000;  // edges
constexpr int D  = 128;     // feature dim

// agg = src (also used to seed agg with x, since eps==0 -> h = x + sum)
__global__ __launch_bounds__(256) void gin_copy_kernel(const float* __restrict__ s,
                                                       float* __restrict__ d) {
  int i = blockIdx.x * blockDim.x + threadIdx.x;
  ((float4*)d)[i] = ((const float4*)s)[i];
}

// one wave32 per edge: lane l moves floats [4l..4l+3] of row x[src[e]] into agg[dst[e]]
__global__ __launch_bounds__(256) void gin_scatter_kernel(const float* __restrict__ x,
                                                          const int* __restrict__ srcIdx,
                                                          const int* __restrict__ dstIdx,
                                                          float* __restrict__ agg) {
  int t = blockIdx.x * blockDim.x + threadIdx.x;
  int e = t >> 5;
  int lane = t & 31;
  if (e >= NE) return;
  int s = srcIdx[e];
  int d = dstIdx[e];
  float4 v = ((const float4*)(x + (size_t)s * D))[lane];
  float* o = agg + (size_t)d * D + lane * 4;
  unsafeAtomicAdd(o + 0, v.x);
  unsafeAtomicAdd(o + 1, v.y);
  unsafeAtomicAdd(o + 2, v.z);
  unsafeAtomicAdd(o + 3, v.w);
}

// out[m][n] = sum_k A[m][k] * W[k][n] + bias[n]  (optional ReLU)
// One wave per 16x16 tile via V_WMMA_F32_16X16X4_F32; block = 16-row strip, 8 waves = 8 N-tiles.
__global__ __launch_bounds__(256) void gin_gemm_kernel(const float* __restrict__ A,
                                                       const float* __restrict__ W,
                                                       const float* __restrict__ bias,
                                                       float* __restrict__ out,
                                                       int relu) {
  int wave = threadIdx.x >> 5;   // 0..7 -> N tile
  int lane = threadIdx.x & 31;
  int half = lane >> 4;          // 0: K=0/1, 1: K=2/3 (A and B layouts, ISA 7.12.2)
  int l16  = lane & 15;
  int mtile = blockIdx.x;        // 0..624

  const float* arow = A + (size_t)(mtile * 16 + l16) * D + 2 * half;
  const float* wcol = W + (size_t)(2 * half) * D + wave * 16 + l16;

  v8f acc = {};
#pragma unroll
  for (int k = 0; k < D; k += 4) {
    v2f a;
    a.x = arow[k + 0];
    a.y = arow[k + 1];
    v2f b;
    b.x = wcol[(size_t)k * D];
    b.y = wcol[(size_t)k * D + D];
    // 8 args: (neg_a, A, neg_b, B, c_mod, C, reuse_a, reuse_b)
    acc = __builtin_amdgcn_wmma_f32_16x16x4_f32(false, a, false, b, (short)0, acc,
                                                false, false);
  }

  float bv = bias[wave * 16 + l16];
  // C/D layout: VGPR r -> row (half*8 + r), col = l16
  float* obase = out + (size_t)(mtile * 16 + half * 8) * D + wave * 16 + l16;
#pragma unroll
  for (int r = 0; r < 8; ++r) {
    float v = acc[r] + bv;
    if (relu) v = fmaxf(v, 0.0f);
    obase[(size_t)r * D] = v;
  }
}

// one wave32 per row, 4 elems/lane, shfl_xor tree reductions (wave32)
__global__ __launch_bounds__(256) void gin_logsoftmax_kernel(float* __restrict__ h) {
  int t = blockIdx.x * blockDim.x + threadIdx.x;
  int row = t >> 5;
  int lane = t & 31;
  if (row >= NN) return;
  float4 v = ((const float4*)(h + (size_t)row * D))[lane];
  float m = fmaxf(fmaxf(v.x, v.y), fmaxf(v.z, v.w));
#pragma unroll
  for (int off = 16; off > 0; off >>= 1) m = fmaxf(m, __shfl_xor(m, off, 32));
  float s = expf(v.x - m) + expf(v.y - m) + expf(v.z - m) + expf(v.w - m);
#pragma unroll
  for (int off = 16; off > 0; off >>= 1) s += __shfl_xor(s, off, 32);
  float ls = m + logf(s);
  v.x -= ls; v.y -= ls; v.z -= ls; v.w -= ls;
  ((float4*)(h + (size_t)row * D))[lane] = v;
}

extern "C" void kernel_launch(void* const* d_in, const int* in_sizes, int n_in,
                              void* d_out, int out_size, void* d_ws, size_t ws_size,
                              hipStream_t stream) {
  const float* x  = (const float*)d_in[0];
  const int*   ei = (const int*)d_in[1];   // [2, E] row-major: src then dst
  const float* w1 = (const float*)d_in[2];
  const float* b1 = (const float*)d_in[3];
  const float* w2 = (const float*)d_in[4];
  const float* b2 = (const float*)d_in[5];
  float* out = (float*)d_out;

  float* agg = (float*)d_ws;               // [NN, D]
  float* h1  = agg + (size_t)NN * D;       // [NN, D]
  const int* srcIdx = ei;
  const int* dstIdx = ei + NE;

  dim3 blk(256);
  const int copyBlocks = (NN * D / 4) / 256;    // 1250
  const int scatBlocks = (NE * 32) / 256;       // 80000
  const int gemmBlocks = NN / 16;               // 625
  const int lsBlocks   = (NN * 32) / 256;       // 1250

  // Layer 1: agg = x + scatter_sum(x) ; h1 = relu(agg @ w1 + b1)
  gin_copy_kernel<<<copyBlocks, blk, 0, stream>>>(x, agg);
  gin_scatter_kernel<<<scatBlocks, blk, 0, stream>>>(x, srcIdx, dstIdx, agg);
  gin_gemm_kernel<<<gemmBlocks, blk, 0, stream>>>(agg, w1, b1, h1, 1);

  // Layer 2: agg = h1 + scatter_sum(h1) ; out = agg @ w2 + b2
  gin_copy_kernel<<<copyBlocks, blk, 0, stream>>>(h1, agg);
  gin_scatter_kernel<<<scatBlocks, blk, 0, stream>>>(h1, srcIdx, dstIdx, agg);
  gin_gemm_kernel<<<gemmBlocks, blk, 0, stream>>>(agg, w2, b2, out, 0);

  // log_softmax in place on d_out
  gin_logsoftmax_kernel<<<lsBlocks, blk, 0, stream>>>(out);
}